// External_attention_79061757985027
// MI455X (gfx1250) — compile-verified
//
#include <hip/hip_runtime.h>
#include <hip/hip_bf16.h>

typedef __attribute__((ext_vector_type(2))) float v2f;
typedef __attribute__((ext_vector_type(8))) float v8f;
typedef __attribute__((ext_vector_type(4))) unsigned int u32x4;
typedef __attribute__((ext_vector_type(4))) int i32x4;
typedef __attribute__((ext_vector_type(8))) int i32x8;

#ifndef __has_builtin
#define __has_builtin(x) 0
#endif
#if __has_builtin(__builtin_amdgcn_tensor_load_to_lds) && \
    __has_builtin(__builtin_amdgcn_s_wait_tensorcnt)
#define HAVE_TDM 1
#else
#define HAVE_TDM 0
#endif

#define EPI_NONE     0
#define EPI_BIAS     1
#define EPI_COLSCALE 2
#define EPI_CONV2    3

#if HAVE_TDM
// ---------------------------------------------------------------------------
// Issue one TDM 2D tile load: Global (tile_d1 rows x tile_d0 elems, row stride
// stride_d0 elems, fp32) -> LDS at lds_byte_addr, with hardware row padding
// (pad_amount dwords appended after every pad_interval dwords).
// D# layout per CDNA5 ISA.
// ---------------------------------------------------------------------------
__device__ __forceinline__ void tdm_load_2d(unsigned lds_byte_addr,
                                            const float* gptr,
                                            int tile_d0, int tile_d1,
                                            int stride_d0,
                                            int pad_interval_code,
                                            int pad_amount_code)
{
    unsigned long long ga = (unsigned long long)(const void*)gptr;
    u32x4 g0;
    g0[0] = 1u;                                   // count=1, user-mode
    g0[1] = lds_byte_addr;                        // lds_addr
    g0[2] = (unsigned)ga;                         // global_addr[31:0]
    g0[3] = (unsigned)(ga >> 32) | (2u << 30);    // global_addr[56:32] | type=2

    i32x8 g1 = {};
    g1[0] = (2 << 16)                             // data_size = 4 bytes
          | (1 << 20)                             // pad_enable
          | (pad_interval_code << 22)             // dwords between pads
          | (pad_amount_code << 25);              // pad dwords - 1
    g1[1] = 0;                                    // no atomic barrier; dim0[15:0]=0
    g1[2] = 0x4000;                               // tensor_dim0 = 1<<30 (no OOB)
    g1[3] = 0x4000 | (tile_d0 << 16);             // tensor_dim1 = 1<<30 | tile_dim0
    g1[4] = tile_d1;                              // tile_dim1 (tile_dim2 = 0)
    g1[5] = stride_d0;                            // tensor_dim0_stride[31:0]
    g1[6] = 0;                                    // stride hi / dim1_stride (unused)
    g1[7] = 0;

    i32x4 z4 = {};
#if __clang_major__ >= 23
    i32x8 z8 = {};
    __builtin_amdgcn_tensor_load_to_lds(g0, g1, z4, z4, z8, 0);
#else
    __builtin_amdgcn_tensor_load_to_lds(g0, g1, z4, z4, 0);
#endif
}
#endif // HAVE_TDM

// ---------------------------------------------------------------------------
// WMMA fp32 GEMM:  C[b] = A (MxK) * B[b] (KxN)  with fused epilogue.
// 256 threads = 8 waves (wave32).  Block tile BM x BN, wave tile (BM/2)x(BN/4).
// K staged through LDS in chunks of 16 (4 wmma k-steps), double buffered.
// LDS staging via Tensor Data Mover (wave 0 drives, TENSORcnt + barrier sync),
// falling back to per-thread loads if the TDM builtin is unavailable.
// ---------------------------------------------------------------------------
template<int BM, int BN, int EPI>
__global__ __launch_bounds__(256) void gemm_wmma_f32(
    const float* __restrict__ A,      // M x K row-major (weights, shared over batch)
    const float* __restrict__ Bm,     // batch-strided K x N row-major
    float*       __restrict__ Cm,     // batch-strided M x N row-major
    const float* __restrict__ extra,  // bias[M] | colsum[b*N+n] | residual x[b*M*N]
    int M, int N, int Kdim)
{
    constexpr int WROWS = 2, WCOLS = 4;
    constexpr int WT_M = BM / WROWS;          // 64 (or 32)
    constexpr int WT_N = BN / WCOLS;          // 32
    constexpr int MT = WT_M / 16;
    constexpr int NT = WT_N / 16;
    constexpr int AP = 20;                    // A LDS pitch: 16 + 4 pad (bank spread)
    constexpr int BP = BN + 16;               // B LDS pitch: 2*144 % 64 == 32 -> the
                                              // two K-row halves hit disjoint banks
    __shared__ float As[2][BM][AP];
    __shared__ float Bs[2][16][BP];

    const int tid = threadIdx.x;
    const int b   = blockIdx.z;
    const int m0  = blockIdx.y * BM;
    const int n0  = blockIdx.x * BN;

    const float* Bb = Bm + (size_t)b * Kdim * N;
    float*       Cb = Cm + (size_t)b * M * N;

    const int wave  = tid >> 5;
    const int lane  = tid & 31;
    const int half  = lane >> 4;              // 0: lanes 0-15, 1: lanes 16-31
    const int l16   = lane & 15;
    const int kbase = half * 2;               // A/B K-sub-offset per ISA layout
    const int wrow  = wave % WROWS;
    const int wcol  = wave / WROWS;

    v8f acc[MT][NT] = {};

    const int Ksteps = Kdim >> 4;

    auto issue_stage = [&](int ks, int buf) {
        const int k0 = ks * 16;
#if HAVE_TDM
        if (wave == 0) {
            // A tile: BM rows x 16 floats; LDS rows padded 16 dw + 4 dw
            tdm_load_2d((unsigned)(size_t)(void*)&As[buf][0][0],
                        A + (size_t)m0 * Kdim + k0,
                        16, BM, Kdim,
                        /*interval 16 dw*/ 3, /*pad 4 dw*/ 3);
            // B tile: 16 rows x BN floats; LDS rows padded BN dw + 16 dw
            tdm_load_2d((unsigned)(size_t)(void*)&Bs[buf][0][0],
                        Bb + (size_t)k0 * N + n0,
                        BN, 16, N,
                        /*interval BN dw*/ __builtin_ctz(BN) - 1, /*pad 16 dw*/ 15);
        }
#else
        // A tile: BM x 16 (row-major, 4-float vectors)
        #pragma unroll
        for (int e = tid; e < BM * 4; e += 256) {
            const int row = e >> 2;
            const int col = (e & 3) << 2;
            const float4 v = *(const float4*)(A + (size_t)(m0 + row) * Kdim + k0 + col);
            *(float4*)&As[buf][row][col] = v;
        }
        // B tile: 16 x BN (coalesced along n)
        #pragma unroll
        for (int e = tid; e < 4 * BN; e += 256) {
            const int r = (e << 2) / BN;
            const int c = (e << 2) % BN;
            const float4 v = *(const float4*)(Bb + (size_t)(k0 + r) * N + n0 + c);
            *(float4*)&Bs[buf][r][c] = v;
        }
#endif
    };

    auto wait_stage = [&]() {
#if HAVE_TDM
        if (wave == 0) __builtin_amdgcn_s_wait_tensorcnt(0);  // DMA done (wave 0)
#endif
        __syncthreads();                                      // publish LDS to all waves
    };

    issue_stage(0, 0);
    wait_stage();

    for (int kt = 0; kt < Ksteps; ++kt) {
        const int cur = kt & 1;
        if (kt + 1 < Ksteps) issue_stage(kt + 1, cur ^ 1);

        #pragma unroll
        for (int kk = 0; kk < 16; kk += 4) {
            v2f af[MT];
            v2f bf[NT];
            // A 16x4 frag: lane holds row M=l16; v.x=K(kbase), v.y=K(kbase+1)
            #pragma unroll
            for (int mt = 0; mt < MT; ++mt) {
                const int row = wrow * WT_M + mt * 16 + l16;
                af[mt].x = As[cur][row][kk + kbase];
                af[mt].y = As[cur][row][kk + kbase + 1];
            }
            // B 4x16 frag: v0 holds K=0/K=2 (by lane half), v1 holds K=1/K=3
            #pragma unroll
            for (int nt = 0; nt < NT; ++nt) {
                const int col = wcol * WT_N + nt * 16 + l16;
                bf[nt].x = Bs[cur][kk + kbase][col];
                bf[nt].y = Bs[cur][kk + kbase + 1][col];
            }
            #pragma unroll
            for (int mt = 0; mt < MT; ++mt)
                #pragma unroll
                for (int nt = 0; nt < NT; ++nt)
                    acc[mt][nt] = __builtin_amdgcn_wmma_f32_16x16x4_f32(
                        false, af[mt], false, bf[nt],
                        (short)0, acc[mt][nt], false, false);
        }
        wait_stage();
    }

    // Store with fused epilogue. C layout: VGPR i -> M = i + 8*half, N = l16.
    #pragma unroll
    for (int mt = 0; mt < MT; ++mt) {
        #pragma unroll
        for (int nt = 0; nt < NT; ++nt) {
            const int colg = n0 + wcol * WT_N + nt * 16 + l16;
            float scale = 1.0f;
            if (EPI == EPI_COLSCALE)
                scale = 1.0f / (1e-9f + extra[(size_t)b * N + colg]);
            #pragma unroll
            for (int i = 0; i < 8; ++i) {
                const int rowg = m0 + wrow * WT_M + mt * 16 + half * 8 + i;
                float v = acc[mt][nt][i];
                if (EPI == EPI_BIAS)     v += extra[rowg];
                if (EPI == EPI_COLSCALE) v *= scale;
                if (EPI == EPI_CONV2) {
                    v = fmaxf(v, 0.0f);                                      // relu(conv2)
                    v += extra[(size_t)b * M * N + (size_t)rowg * N + colg]; // + idn
                    v = fmaxf(v, 0.0f);                                      // final relu
                }
                Cb[(size_t)rowg * N + colg] = v;
            }
        }
    }
}

// ---------------------------------------------------------------------------
// Softmax over the token axis (length 4096) for each (b,k) row of attn.
// ---------------------------------------------------------------------------
__global__ __launch_bounds__(256) void softmax_rows(float* __restrict__ attn, int N)
{
    float* p = attn + (size_t)blockIdx.x * N;
    const int tid = threadIdx.x;
    __shared__ float red[256];

    float v[16];
    float m = -3.402823466e+38f;
    #pragma unroll
    for (int i = 0; i < 16; ++i) {
        v[i] = p[tid + i * 256];
        m = fmaxf(m, v[i]);
    }
    red[tid] = m; __syncthreads();
    for (int s = 128; s > 0; s >>= 1) {
        if (tid < s) red[tid] = fmaxf(red[tid], red[tid + s]);
        __syncthreads();
    }
    const float mx = red[0];
    __syncthreads();

    float sum = 0.0f;
    #pragma unroll
    for (int i = 0; i < 16; ++i) {
        v[i] = __expf(v[i] - mx);
        sum += v[i];
    }
    red[tid] = sum; __syncthreads();
    for (int s = 128; s > 0; s >>= 1) {
        if (tid < s) red[tid] += red[tid + s];
        __syncthreads();
    }
    const float inv = 1.0f / red[0];
    #pragma unroll
    for (int i = 0; i < 16; ++i)
        p[tid + i * 256] = v[i] * inv;
}

// ---------------------------------------------------------------------------
// Per-(b,n) sum over the latent dim k (64) of softmaxed attn -> colsum[b*N+n].
// (The L1 renorm is applied exactly inside lin1's column-scale epilogue.)
// ---------------------------------------------------------------------------
__global__ __launch_bounds__(256) void colsum_kernel(const float* __restrict__ attn,
                                                     float* __restrict__ colsum,
                                                     int K, int N)
{
    const int idx = blockIdx.x * 256 + threadIdx.x;   // b*N + n
    const int b = idx / N;
    const int n = idx % N;
    const float* p = attn + (size_t)b * K * N + n;
    float s = 0.0f;
    #pragma unroll 8
    for (int k = 0; k < K; ++k)
        s += p[(size_t)k * N];
    colsum[idx] = s;
}

// ---------------------------------------------------------------------------
// torch .view(-1, c) rank op: each row = 512 CONSECUTIVE flat elements of
// (b,c,n) row-major y, i.e. 512 consecutive n for a fixed channel.
// thr = 256th largest (= ascending-sorted[256]); <thr -> 0.75x else 1.25x.
// Bitonic sort of 512 floats in LDS with 256 threads; scale in place.
// ---------------------------------------------------------------------------
__global__ __launch_bounds__(256) void rank_rows(float* __restrict__ y)
{
    __shared__ float s[512];
    const size_t base = (size_t)blockIdx.x * 512;
    const int tid = threadIdx.x;

    const float o0 = y[base + tid];
    const float o1 = y[base + tid + 256];
    s[tid] = o0;
    s[tid + 256] = o1;
    __syncthreads();

    for (int k = 2; k <= 512; k <<= 1) {
        for (int j = k >> 1; j > 0; j >>= 1) {
            #pragma unroll
            for (int t = 0; t < 2; ++t) {
                const int i = tid + t * 256;
                const int ixj = i ^ j;
                if (ixj > i) {
                    const bool up = ((i & k) == 0);
                    const float a = s[i], c = s[ixj];
                    if ((a > c) == up) { s[i] = c; s[ixj] = a; }
                }
            }
            __syncthreads();
        }
    }
    const float thr = s[256];                 // k-th largest, k = c/2 = 256
    y[base + tid]       = (o0 < thr) ? 0.75f * o0 : 1.25f * o0;
    y[base + tid + 256] = (o1 < thr) ? 0.75f * o1 : 1.25f * o1;
}

// ---------------------------------------------------------------------------
extern "C" void kernel_launch(void* const* d_in, const int* in_sizes, int n_in,
                              void* d_out, int out_size, void* d_ws, size_t ws_size,
                              hipStream_t stream)
{
    const float* x     = (const float*)d_in[0];   // (16,512,64,64)
    const float* w1    = (const float*)d_in[1];   // (512,512)
    const float* b1    = (const float*)d_in[2];   // (512,)
    const float* lin0w = (const float*)d_in[3];   // (64,512)
    const float* lin1w = (const float*)d_in[4];   // (512,64)
    const float* w2    = (const float*)d_in[5];   // (512,512)
    float* out = (float*)d_out;

    const int Bc = 16, C = 512, K = 64, N = 4096;

    float* Y      = (float*)d_ws;                    // B*C*N  (134 MB)
    float* attn   = Y + (size_t)Bc * C * N;          // B*K*N  (17 MB)
    float* colsum = attn + (size_t)Bc * K * N;       // B*N

    dim3 blk(256);

    // conv1: Y = W1 * x + b1
    gemm_wmma_f32<128, 128, EPI_BIAS>
        <<<dim3(N / 128, C / 128, Bc), blk, 0, stream>>>(w1, x, Y, b1, C, N, C);

    // lin0: attn_logits = L0 * Y
    gemm_wmma_f32<64, 128, EPI_NONE>
        <<<dim3(N / 128, 1, Bc), blk, 0, stream>>>(lin0w, Y, attn, nullptr, K, N, C);

    // softmax over tokens, in place
    softmax_rows<<<dim3(Bc * K), blk, 0, stream>>>(attn, N);

    // per-(b,n) sum over k for the L1 renorm
    colsum_kernel<<<dim3(Bc * N / 256), blk, 0, stream>>>(attn, colsum, K, N);

    // lin1: Y = (L1 * attn) / (1e-9 + colsum)   (overwrites Y)
    gemm_wmma_f32<128, 128, EPI_COLSCALE>
        <<<dim3(N / 128, C / 128, Bc), blk, 0, stream>>>(lin1w, attn, Y, colsum, C, N, K);

    // rank / top-k scaling in place (65536 rows of 512)
    rank_rows<<<dim3((unsigned)((size_t)Bc * C * N / 512)), blk, 0, stream>>>(Y);

    // conv2 + relu + residual + relu -> out
    gemm_wmma_f32<128, 128, EPI_CONV2>
        <<<dim3(N / 128, C / 128, Bc), blk, 0, stream>>>(w2, Y, out, x, C, N, C);
}